// GroupedQueryAttentionAdapter_47244640256476
// MI455X (gfx1250) — compile-verified
//
#include <hip/hip_runtime.h>

typedef __attribute__((ext_vector_type(2))) float v2f;
typedef __attribute__((ext_vector_type(4))) float v4f;
typedef __attribute__((ext_vector_type(8))) float v8f;

#define QUERY_HEADS 16
#define HEAD_DIM 128
#define KV_HEADS 4
#define QPK 4
#define LOCAL_WINDOW 4096
#define B_ 8
#define S_ 8
#define MODEL_DIM 2048
#define CTX 8192
#define WIN_START (CTX - LOCAL_WINDOW)
#define SPLITS 16
#define KCHUNK (LOCAL_WINDOW / SPLITS) /* 256 */
#define NBLK (KCHUNK / 16)             /* 16 key blocks per split */
#define EPS 1e-6f
#define NEG_HUGE (-3.402823466e38f)

static __device__ __forceinline__ v8f wmma4(v2f a, v2f b, v8f c) {
  // D = A(16x4,f32) * B(4x16,f32) + C(16x16,f32); exact f32 matrix op on CDNA5
  return __builtin_amdgcn_wmma_f32_16x16x4_f32(false, a, false, b, (short)0, c,
                                               false, false);
}

// ---- CDNA5 async global->LDS path (ASYNCcnt), with graceful fallback -------
#if __has_builtin(__builtin_amdgcn_global_load_async_to_lds_b128) && \
    __has_builtin(__builtin_amdgcn_s_wait_asynccnt)
#define HAVE_ASYNC_LDS 1
#else
#define HAVE_ASYNC_LDS 0
#endif

static __device__ __forceinline__ void lds_copy_b128(const float* gsrc,
                                                     float* ldst) {
#if HAVE_ASYNC_LDS
  // Builtin expects: (int4 __device__*, int4 __shared__*, imm offset, imm cpol)
  // with GCC vector_size(16) int4 pointees (per hipcc diagnostic).
  typedef int v4i_gcc __attribute__((vector_size(16)));
  typedef __attribute__((address_space(1))) v4i_gcc gv4i;
  typedef __attribute__((address_space(3))) v4i_gcc sv4i;
  __builtin_amdgcn_global_load_async_to_lds_b128((gv4i*)gsrc, (sv4i*)ldst, 0,
                                                 0);
#else
  *(v4f*)ldst = *(const v4f*)gsrc;
#endif
}

#if HAVE_ASYNC_LDS
#define WAIT_ASYNC(n) __builtin_amdgcn_s_wait_asynccnt(n)
#else
#define WAIT_ASYNC(n) ((void)0)
#endif

// ---------------------------------------------------------------------------
// C[m][n] = sum_k A[m][k] * W[n][k]   (A @ W^T), one wave per 16x16 tile.
// A fragment: lane l -> m = l&15, k pair at (l>>4)*2 (8B aligned -> b64 loads)
// B fragment: lane l -> n = l&15 (row of W), same k pair.
// D: vgpr i, lane l -> m = i + 8*(l>>4), n = l&15.
// ---------------------------------------------------------------------------
__global__ __launch_bounds__(32) void gemm_xwT_wmma(
    const float* __restrict__ A, const float* __restrict__ W,
    float* __restrict__ C, int M, int N, int K) {
  const int lane = threadIdx.x & 31;
  const int tiles_n = N >> 4;
  const int mt = blockIdx.x / tiles_n;
  const int nt = blockIdx.x % tiles_n;
  const int r = lane & 15;
  const int g = lane >> 4;
  const int kh = g << 1;

  const float* arow = A + (size_t)(mt * 16 + r) * K + kh;
  const float* wrow = W + (size_t)(nt * 16 + r) * K + kh;

  v8f acc = {};
#pragma unroll 4
  for (int k0 = 0; k0 < K; k0 += 4) {
    v2f a = *(const v2f*)(arow + k0);
    v2f b = *(const v2f*)(wrow + k0);
    acc = wmma4(a, b, acc);
  }

  float* crow = C + (size_t)(mt * 16 + 8 * g) * N + nt * 16 + r;
#pragma unroll
  for (int i = 0; i < 8; ++i) crow[(size_t)i * N] = acc[i];
}

// ---------------------------------------------------------------------------
// RMS-norm each (row, head) 128-vector of qproj, scale by qnorm_w, and
// rearrange into qattn[b][kvh][s*QPK + j][d].
// ---------------------------------------------------------------------------
__global__ __launch_bounds__(128) void qnorm_rearrange(
    const float* __restrict__ qproj, const float* __restrict__ qnorm_w,
    float* __restrict__ qattn) {
  const int row = blockIdx.x >> 4;   // b*S + s  in [0,64)
  const int head = blockIdx.x & 15;  // query head
  const int d = threadIdx.x;

  float x = qproj[(size_t)row * (QUERY_HEADS * HEAD_DIM) + head * HEAD_DIM + d];
  float ss = x * x;
#pragma unroll
  for (int m = 16; m >= 1; m >>= 1) ss += __shfl_xor(ss, m, 32);

  __shared__ float part[4];
  if ((threadIdx.x & 31) == 0) part[threadIdx.x >> 5] = ss;
  __syncthreads();
  float tot = part[0] + part[1] + part[2] + part[3];
  float inv = rsqrtf(tot * (1.0f / HEAD_DIM) + EPS);
  float y = x * inv * qnorm_w[d];

  const int b = row / S_, s = row % S_;
  const int h = head / QPK, j = head % QPK;
  qattn[(((size_t)(b * KV_HEADS + h) * 32) + (s * QPK + j)) * HEAD_DIM + d] = y;
}

// ---------------------------------------------------------------------------
// Flash attention, split across the 4096-token window.
// grid = (B*KV_HEADS) * SPLITS blocks, 2 waves per block (one 16-row q-tile
// each). K/V tiles stream global->LDS via async b128 copies, double-buffered
// (ASYNCcnt). Q fragments live in registers for the whole split. Online
// softmax per 16-key block; P tile routed through LDS for the D->A layout
// transpose; unnormalized partial (O, m, l) written per split.
// ---------------------------------------------------------------------------
__global__ __launch_bounds__(64) void attn_flash_split(
    const float* __restrict__ qattn, const float* __restrict__ kcache,
    const float* __restrict__ vcache, const unsigned char* __restrict__ mask,
    float* __restrict__ part_o, float* __restrict__ part_ml) {
  const int split = blockIdx.x % SPLITS;
  const int bh = blockIdx.x / SPLITS;  // b*KV_HEADS + h
  const int b = bh / KV_HEADS;
  const int tid = threadIdx.x;
  const int lane = tid & 31;
  const int wave = tid >> 5;
  const int qbase = wave * 16;
  const int r = lane & 15;
  const int g = lane >> 4;
  const int kh = g << 1;

  const float* Q = qattn + (size_t)bh * 32 * HEAD_DIM;
  const int kstart = WIN_START + split * KCHUNK;
  const float* Kp = kcache + ((size_t)bh * CTX + kstart) * HEAD_DIM;
  const float* Vp = vcache + ((size_t)bh * CTX + kstart) * HEAD_DIM;
  const unsigned char* Mp = mask + ((size_t)b * 32) * CTX + kstart;

  __shared__ alignas(16) float Kbuf[2][16 * HEAD_DIM];  // 8 KB each
  __shared__ alignas(16) float Vbuf[2][16 * HEAD_DIM];  // 8 KB each
  __shared__ float Pbuf[2][16 * 16];
  float* P = Pbuf[wave];

  // Q fragments for this wave's 16 rows: resident in VGPRs for the whole split
  const float* qrow = Q + (size_t)(qbase + r) * HEAD_DIM + kh;
  v2f qfrag[HEAD_DIM / 4];
#pragma unroll
  for (int k = 0; k < HEAD_DIM / 4; ++k)
    qfrag[k] = *(const v2f*)(qrow + 4 * k);

  float row_max[8], row_sum[8];
#pragma unroll
  for (int i = 0; i < 8; ++i) {
    row_max[i] = NEG_HUGE;
    row_sum[i] = 0.0f;
  }
  v8f oacc[8] = {};

  // Issue the K/V tile copy for key block `kb16` into buffer `buf`.
  // 16 rows x 128 floats = 512 b128 granules; 64 threads x 8 granules each.
  // Per wave: 16 async instructions (8 K + 8 V).
  auto issue_tile = [&](int kb16, int buf) {
    const float* gK = Kp + (size_t)kb16 * HEAD_DIM;
    const float* gV = Vp + (size_t)kb16 * HEAD_DIM;
#pragma unroll
    for (int i = 0; i < 8; ++i) {
      const int e = (tid + 64 * i) * 4;
      lds_copy_b128(gK + e, &Kbuf[buf][e]);
      lds_copy_b128(gV + e, &Vbuf[buf][e]);
    }
  };

  issue_tile(0, 0);

  for (int nb = 0; nb < NBLK; ++nb) {
    const int cur = nb & 1;
    const int kb = nb * 16;
    if (nb + 1 < NBLK) {
      issue_tile(kb + 16, cur ^ 1);  // prefetch next tile into other buffer
      WAIT_ASYNC(16);                // previous 16 (current tile) complete
    } else {
      WAIT_ASYNC(0);
    }
    __syncthreads();  // current K/V tile visible to both waves

    // S = Q (16x128) @ K^T (128x16), K fragments from LDS
    const float* kf = &Kbuf[cur][(size_t)r * HEAD_DIM + kh];
    v8f sacc = {};
#pragma unroll 8
    for (int k = 0; k < HEAD_DIM / 4; ++k)
      sacc = wmma4(qfrag[k], *(const v2f*)(kf + 4 * k), sacc);

    // Mask + online softmax. Element (i,g) is row m=i+8g, col n=r.
    float p[8], csc[8];
#pragma unroll
    for (int i = 0; i < 8; ++i) {
      const int qg = qbase + i + 8 * g;
      const unsigned char mk = Mp[(size_t)qg * CTX + kb + r];
      float s = mk ? sacc[i] : NEG_HUGE;
      float mx = s;
      mx = fmaxf(mx, __shfl_xor(mx, 1, 32));
      mx = fmaxf(mx, __shfl_xor(mx, 2, 32));
      mx = fmaxf(mx, __shfl_xor(mx, 4, 32));
      mx = fmaxf(mx, __shfl_xor(mx, 8, 32));
      float mn = fmaxf(row_max[i], mx);
      float pe = (s == NEG_HUGE) ? 0.0f : __expf(s - mn);
      float sum = pe;
      sum += __shfl_xor(sum, 1, 32);
      sum += __shfl_xor(sum, 2, 32);
      sum += __shfl_xor(sum, 4, 32);
      sum += __shfl_xor(sum, 8, 32);
      float sc = __expf(row_max[i] - mn);
      row_sum[i] = row_sum[i] * sc + sum;
      row_max[i] = mn;
      csc[i] = sc;
      p[i] = pe;
    }
#pragma unroll
    for (int nd = 0; nd < 8; ++nd)
#pragma unroll
      for (int i = 0; i < 8; ++i) oacc[nd][i] *= csc[i];

    // P tile: D-layout -> LDS row-major -> reload as A fragments
    __syncthreads();
#pragma unroll
    for (int i = 0; i < 8; ++i) P[(i + 8 * g) * 16 + r] = p[i];
    __syncthreads();

    // O += P (16x16) @ V (16x128), V fragments from LDS
#pragma unroll
    for (int nd = 0; nd < 8; ++nd) {
      const float* vcol = &Vbuf[cur][nd * 16 + r];
#pragma unroll
      for (int kk = 0; kk < 4; ++kk) {
        const int k0 = kk * 4 + kh;
        v2f a = *(const v2f*)(P + r * 16 + k0);
        v2f bv;
        bv.x = vcol[(size_t)k0 * HEAD_DIM];
        bv.y = vcol[(size_t)(k0 + 1) * HEAD_DIM];
        oacc[nd] = wmma4(a, bv, oacc[nd]);
      }
    }
    __syncthreads();  // all reads of buffer `cur` done before it is re-issued
  }

  // Write unnormalized partial O and (m, l) stats for this split
  float* op = part_o + ((size_t)bh * SPLITS + split) * 32 * HEAD_DIM;
#pragma unroll
  for (int nd = 0; nd < 8; ++nd)
#pragma unroll
    for (int i = 0; i < 8; ++i)
      op[(size_t)(qbase + i + 8 * g) * HEAD_DIM + nd * 16 + r] = oacc[nd][i];

  float* mlp = part_ml + ((size_t)bh * SPLITS + split) * 64;  // [m x32 | l x32]
#pragma unroll
  for (int i = 0; i < 8; ++i) {
    if (r == i) {
      mlp[qbase + i + 8 * g] = row_max[i];
      mlp[32 + qbase + i + 8 * g] = row_sum[i];
    }
  }
}

// ---------------------------------------------------------------------------
// LSE-merge the split partials, normalize, and write context in (b, s, 2048)
// layout ready for the o-projection GEMM.
// ---------------------------------------------------------------------------
__global__ __launch_bounds__(128) void attn_combine(
    const float* __restrict__ part_o, const float* __restrict__ part_ml,
    float* __restrict__ ctxbuf) {
  const int bh = blockIdx.x >> 5;
  const int qrow = blockIdx.x & 31;
  const int d = threadIdx.x;
  const int b = bh / KV_HEADS, h = bh % KV_HEADS;

  float M = NEG_HUGE;
  for (int sp = 0; sp < SPLITS; ++sp)
    M = fmaxf(M, part_ml[((size_t)bh * SPLITS + sp) * 64 + qrow]);

  float L = 0.0f, O = 0.0f;
  for (int sp = 0; sp < SPLITS; ++sp) {
    const float* mlp = part_ml + ((size_t)bh * SPLITS + sp) * 64;
    float m = mlp[qrow], l = mlp[32 + qrow];
    float w = (m > -1e37f) ? __expf(m - M) : 0.0f;
    L += l * w;
    O += part_o[(((size_t)bh * SPLITS + sp) * 32 + qrow) * HEAD_DIM + d] * w;
  }
  float out = (L > 0.0f) ? (O / L) : 0.0f;

  const int s = qrow / QPK, j = qrow % QPK;
  ctxbuf[(size_t)(b * S_ + s) * (QUERY_HEADS * HEAD_DIM) +
         (h * QPK + j) * HEAD_DIM + d] = out;
}

extern "C" void kernel_launch(void* const* d_in, const int* in_sizes, int n_in,
                              void* d_out, int out_size, void* d_ws,
                              size_t ws_size, hipStream_t stream) {
  const float* hidden = (const float*)d_in[0];
  const float* kcache = (const float*)d_in[1];
  const float* vcache = (const float*)d_in[2];
  const unsigned char* mask = (const unsigned char*)d_in[3];
  const float* q_proj_w = (const float*)d_in[4];
  const float* o_proj_w = (const float*)d_in[5];
  const float* qnorm_w = (const float*)d_in[6];
  float* out = (float*)d_out;

  float* ws = (float*)d_ws;
  float* qproj = ws;                                      // 64*2048
  float* qattn = qproj + 64 * 2048;                       // 32*32*128
  float* ctxbuf = qattn + 32 * 32 * HEAD_DIM;             // 64*2048
  float* part_o = ctxbuf + 64 * 2048;                     // 32*SPLITS*32*128
  float* part_ml = part_o + 32 * SPLITS * 32 * HEAD_DIM;  // 32*SPLITS*64

  const int M = B_ * S_;  // 64
  // 1) Q projection: (64x2048) @ (2048x2048)^T
  gemm_xwT_wmma<<<dim3((M / 16) * (MODEL_DIM / 16)), dim3(32), 0, stream>>>(
      hidden, q_proj_w, qproj, M, QUERY_HEADS * HEAD_DIM, MODEL_DIM);
  // 2) RMS-norm + head rearrange
  qnorm_rearrange<<<dim3(M * QUERY_HEADS), dim3(128), 0, stream>>>(
      qproj, qnorm_w, qattn);
  // 3) split flash attention over 4096-token window
  attn_flash_split<<<dim3(B_ * KV_HEADS * SPLITS), dim3(64), 0, stream>>>(
      qattn, kcache, vcache, mask, part_o, part_ml);
  // 4) merge splits -> context
  attn_combine<<<dim3(B_ * KV_HEADS * 32), dim3(128), 0, stream>>>(
      part_o, part_ml, ctxbuf);
  // 5) O projection: (64x2048) @ (2048x2048)^T -> output
  gemm_xwT_wmma<<<dim3((M / 16) * (MODEL_DIM / 16)), dim3(32), 0, stream>>>(
      ctxbuf, o_proj_w, out, M, MODEL_DIM, QUERY_HEADS * HEAD_DIM);
}